// yoloLoss_21818433864438
// MI455X (gfx1250) — compile-verified
//
#include <hip/hip_runtime.h>

typedef __attribute__((ext_vector_type(2))) float v2f;
typedef __attribute__((ext_vector_type(8))) float v8f;

#define NCELLS (8192 * 14 * 14)   // 1,605,632

__global__ __launch_bounds__(256) void yolo_loss_main(
    const float* __restrict__ pred, const float* __restrict__ targ,
    double* __restrict__ acc_out) {
  const unsigned stride = gridDim.x * blockDim.x;
  float acc = 0.0f;

  for (unsigned cell = blockIdx.x * blockDim.x + threadIdx.x; cell < NCELLS;
       cell += stride) {
    // Prefetch the next iteration's cell (global_prefetch_b8).
    unsigned nxt = cell + stride;
    if (nxt < NCELLS) {
      __builtin_prefetch(pred + (size_t)nxt * 30, 0, 0);
      __builtin_prefetch(targ + (size_t)nxt * 30, 0, 0);
    }

    // 30 floats per cell = 15 x float2 (8-byte aligned) -> global_load_b64.
    const float2* pp = reinterpret_cast<const float2*>(pred) + (size_t)cell * 15;
    const float2* tp = reinterpret_cast<const float2*>(targ) + (size_t)cell * 15;
    float p[30], t[30];
#pragma unroll
    for (int k = 0; k < 15; ++k) {
      float2 a = pp[k];
      float2 b = tp[k];
      p[2 * k] = a.x; p[2 * k + 1] = a.y;
      t[2 * k] = b.x; t[2 * k + 1] = b.y;
    }

    const float objf = (t[4] > 0.0f) ? 1.0f : 0.0f;
    const float noof = 1.0f - objf;

    // no-object confidence loss (conf indices 4 and 9)
    float d4 = p[4] - t[4];
    float d9 = p[9] - t[9];
    float noobj = noof * (d4 * d4 + d9 * d9);

    // target box (first 5 of target) in xyxy
    const float inv14 = 1.0f / 14.0f;
    float tcx = t[0] * inv14, tcy = t[1] * inv14;
    float tw = t[2], th = t[3];
    float tx0 = tcx - 0.5f * tw, ty0 = tcy - 0.5f * th;
    float tx1 = tcx + 0.5f * tw, ty1 = tcy + 0.5f * th;
    float area_t = (tx1 - tx0) * (ty1 - ty0);

    // IOU of both predicted boxes vs target box
    float iou[2];
#pragma unroll
    for (int b = 0; b < 2; ++b) {
      int o = 5 * b;
      float cx = p[o] * inv14, cy = p[o + 1] * inv14;
      float w = p[o + 2], h = p[o + 3];
      float x0 = cx - 0.5f * w, y0 = cy - 0.5f * h;
      float x1 = cx + 0.5f * w, y1 = cy + 0.5f * h;
      float ltx = fmaxf(x0, tx0), lty = fmaxf(y0, ty0);
      float rbx = fminf(x1, tx1), rby = fminf(y1, ty1);
      float iw = fmaxf(rbx - ltx, 0.0f), ih = fmaxf(rby - lty, 0.0f);
      float inter = iw * ih;
      float area_p = (x1 - x0) * (y1 - y0);
      iou[b] = inter / (area_p + area_t - inter);
    }
    // argmax with first-occurrence tie-break (box 0 on tie)
    int ridx = (iou[1] > iou[0]) ? 1 : 0;
    float max_iou = fmaxf(iou[0], iou[1]);

    float loss = 0.5f * noobj;

    // class loss (channels 10..29)
    float cls = 0.0f;
#pragma unroll
    for (int c = 10; c < 30; ++c) {
      float d = p[c] - t[c];
      cls += d * d;
    }
    loss += objf * cls;

    // per-box responsible / not-responsible terms
#pragma unroll
    for (int b = 0; b < 2; ++b) {
      int o = 5 * b;
      float r = (b == ridx) ? objf : 0.0f;   // resp
      float nr = (b == ridx) ? 0.0f : objf;  // notresp
      float conf = p[o + 4];
      float dc = conf - max_iou;
      loss += 2.0f * r * dc * dc;            // contain
      float dx = p[o] - t[o];
      float dy = p[o + 1] - t[o + 1];
      float sw = sqrtf(p[o + 2]) - sqrtf(t[o + 2]);
      float sh = sqrtf(p[o + 3]) - sqrtf(t[o + 3]);
      loss += 5.0f * r * (dx * dx + dy * dy + sw * sw + sh * sh);  // loc
      loss += nr * conf * conf;              // not_contain
    }

    acc += loss;
  }

  // ---- Wave-level reduction via V_WMMA_F32_16X16X4_F32 ----
  // A = ones(16x4) => D[m][n] = sum_k B[k][n] + C[m][n].
  // Sum over all n of row 0 equals the sum of all 64 B slots regardless of
  // the exact (k,n)<->(lane,vgpr) mapping; B.vgpr1 = 0 so this is the exact
  // f32 sum of the 32 lane partials. Executed by all waves with full EXEC.
  v2f amat; amat.x = 1.0f; amat.y = 1.0f;
  v2f bmat; bmat.x = acc;  bmat.y = 0.0f;
  v8f cmat = {};
  cmat = __builtin_amdgcn_wmma_f32_16x16x4_f32(
      /*neg_a=*/false, amat, /*neg_b=*/false, bmat,
      /*c_mod=*/(short)0, cmat, /*reuse_a=*/false, /*reuse_b=*/false);
  float s = cmat[0];  // D[row0][lane&15] (rows are identical)
  // sum the 16 columns (xor stays within each half-wave group of 16)
  s += __shfl_xor(s, 1, 32);
  s += __shfl_xor(s, 2, 32);
  s += __shfl_xor(s, 4, 32);
  s += __shfl_xor(s, 8, 32);

  __shared__ float swave[8];
  int lane = threadIdx.x & 31;
  int wv = threadIdx.x >> 5;
  if (lane == 0) swave[wv] = s;
  __syncthreads();
  if (threadIdx.x == 0) {
    float tot = 0.0f;
#pragma unroll
    for (int i = 0; i < 8; ++i) tot += swave[i];
    atomicAdd(acc_out, (double)tot);  // global_atomic_add_f64
  }
}

__global__ void yolo_loss_finalize(const double* __restrict__ acc,
                                   float* __restrict__ out) {
  out[0] = (float)(acc[0] * (1.0 / 8192.0));
}

extern "C" void kernel_launch(void* const* d_in, const int* in_sizes, int n_in,
                              void* d_out, int out_size, void* d_ws,
                              size_t ws_size, hipStream_t stream) {
  (void)in_sizes; (void)n_in; (void)out_size; (void)ws_size;
  const float* pred = (const float*)d_in[0];
  const float* targ = (const float*)d_in[1];
  double* acc = (double*)d_ws;

  hipMemsetAsync(acc, 0, sizeof(double), stream);  // graph-capture safe

  const int threads = 256;
  const int blocks = NCELLS / (threads * 2);  // 3136: 2 cells/thread, exact
  yolo_loss_main<<<blocks, threads, 0, stream>>>(pred, targ, acc);
  yolo_loss_finalize<<<1, 1, 0, stream>>>(acc, (float*)d_out);
}